// MoleculeEmbedding_51788715655337
// MI455X (gfx1250) — compile-verified
//
#include <hip/hip_runtime.h>
#include <hip/hip_bf16.h>

// ---------------------------------------------------------------------------
// MoleculeEmbedding on MI455X (gfx1250, wave32, WMMA).
//
// Mathematical collapse (see analysis): softmax over size-1 axis == 1, and all
// per-edge state depends only on dst, so:
//   deg[v]   = #edges with dst==v
//   layer0:  s = sum_k x[v][k];  h = s (broadcast)
//            cs = elu(s*twsum0 + tb0)            (twsum0[j] = sum_k tw0[j][k])
//            gi = cs @ wih0.T + bih0             (real GEMM, K=128, 384 cols)
//            gh = s*whhsum0 + bhh0
//            x1[v] = deg[v] * GRU_combine(gi, gh, h)
//   layer1:  h = x1[v]
//            cs = elu(x1 @ tw1.T + tb1)          (GEMM, 128 cols)
//            gi = cs @ wih1.T + bih1             (GEMM, 384 cols)
//            gh = x1 @ whh1.T + bhh1             (GEMM, 384 cols)
//            out[v] = deg[v] * GRU_combine(gi, gh, h)
// All GEMMs run on V_WMMA_F32_16X16X4_F32, one wave per 16-node tile.
// ---------------------------------------------------------------------------

#define N_NODES 200000
#define N_EDGES 600000
#define FPDIM   128
#define NTILES  (N_NODES / 16)   // 12500

typedef float v8f __attribute__((ext_vector_type(8)));
typedef float v2f __attribute__((ext_vector_type(2)));

__device__ __forceinline__ v8f wmma_f32_k4(v2f a, v2f b, v8f c) {
  // D = A(16x4 f32) x B(4x16 f32) + C(16x16 f32)
  return __builtin_amdgcn_wmma_f32_16x16x4_f32(false, a, false, b, (short)0, c,
                                               false, false);
}

__device__ __forceinline__ float elu1f(float u) {
  return u > 0.0f ? u : expm1f(u);
}
__device__ __forceinline__ float sigm1f(float u) {
  return 1.0f / (1.0f + expf(-u));
}

// --------------------------- small prep kernels ----------------------------

__global__ void zero_i32_kernel(int* p, int n) {
  int i = blockIdx.x * blockDim.x + threadIdx.x;
  if (i < n) p[i] = 0;
}

__global__ void deg_hist_kernel(const int* __restrict__ dst, int* __restrict__ deg, int e) {
  int i = blockIdx.x * blockDim.x + threadIdx.x;
  if (i < e) atomicAdd(&deg[dst[i]], 1);
}

// one wave per row: s[v] = sum_k x[v][k]
__global__ void rowsum_kernel(const float* __restrict__ x, float* __restrict__ s, int n) {
  int row  = (int)((blockIdx.x * blockDim.x + threadIdx.x) >> 5);
  int lane = threadIdx.x & 31;
  if (row >= n) return;
  float4 v = *(const float4*)(x + (size_t)row * FPDIM + 4 * lane);
  float p = v.x + v.y + v.z + v.w;
  for (int off = 16; off; off >>= 1) p += __shfl_xor(p, off, 32);
  if (lane == 0) s[row] = p;
}

// rows 0..127 -> twsum0 (from attend_w0), rows 128..511 -> whhsum0 (from whh0)
__global__ void wsum_kernel(const float* __restrict__ aw, const float* __restrict__ whh,
                            float* __restrict__ twsum, float* __restrict__ whhsum) {
  int row  = (int)((blockIdx.x * blockDim.x + threadIdx.x) >> 5);
  int lane = threadIdx.x & 31;
  if (row >= 512) return;
  const float* src = (row < 128) ? (aw + (size_t)row * FPDIM)
                                 : (whh + (size_t)(row - 128) * FPDIM);
  float4 v = *(const float4*)(src + 4 * lane);
  float p = v.x + v.y + v.z + v.w;
  for (int off = 16; off; off >>= 1) p += __shfl_xor(p, off, 32);
  if (lane == 0) {
    if (row < 128) twsum[row] = p; else whhsum[row - 128] = p;
  }
}

// ------------------------------- layer 0 -----------------------------------
// 4 waves/block, one 16-node tile per wave.
__global__ __launch_bounds__(128) void layer0_kernel(
    const float* __restrict__ sbuf, const int* __restrict__ deg,
    const float* __restrict__ twsum, const float* __restrict__ tb,
    const float* __restrict__ whhsum, const float* __restrict__ wih,
    const float* __restrict__ bih, const float* __restrict__ bhh,
    float* __restrict__ x1) {
  __shared__ float sh_s[4][16];
  __shared__ float sh_d[4][16];
  const int lane = threadIdx.x & 31;
  const int wave = threadIdx.x >> 5;
  const int tile = blockIdx.x * 4 + wave;
  const int node0 = tile * 16;

  if (lane < 16) {
    sh_s[wave][lane] = sbuf[node0 + lane];
    sh_d[wave][lane] = (float)deg[node0 + lane];
  }
  __syncthreads();

  const int m  = lane & 15;            // A: row;  B: col within tile
  const int kh = (lane >> 4) << 1;     // K sub-offset per ISA A/B layout
  const float sm = sh_s[wave][m];

  // A fragments: cs[m][k] = elu(s*twsum[k] + tb[k]); K=128 -> 32 steps of 4
  v2f afrag[32];
#pragma unroll
  for (int t = 0; t < 32; ++t) {
    int k0 = 4 * t + kh;
    v2f tw  = *(const v2f*)(twsum + k0);
    v2f tbv = *(const v2f*)(tb + k0);
    afrag[t].x = elu1f(sm * tw.x + tbv.x);
    afrag[t].y = elu1f(sm * tw.y + tbv.y);
  }

#pragma unroll 1
  for (int c = 0; c < 8; ++c) {
    v8f dr = {0,0,0,0,0,0,0,0}, dz = {0,0,0,0,0,0,0,0}, dn = {0,0,0,0,0,0,0,0};
#pragma unroll
    for (int t = 0; t < 32; ++t) {
      int k0 = 4 * t + kh;
      v2f br = *(const v2f*)(wih + (size_t)(16 * c +       m) * FPDIM + k0);
      v2f bz = *(const v2f*)(wih + (size_t)(16 * c + 128 + m) * FPDIM + k0);
      v2f bn = *(const v2f*)(wih + (size_t)(16 * c + 256 + m) * FPDIM + k0);
      dr = wmma_f32_k4(afrag[t], br, dr);
      dz = wmma_f32_k4(afrag[t], bz, dz);
      dn = wmma_f32_k4(afrag[t], bn, dn);
    }
    const int j = 16 * c + m;
    const float bir = bih[j], biz = bih[j + 128], bin = bih[j + 256];
    const float whr = whhsum[j], whz = whhsum[j + 128], whn = whhsum[j + 256];
    const float bhr = bhh[j], bhz = bhh[j + 128], bhn = bhh[j + 256];
#pragma unroll
    for (int i = 0; i < 8; ++i) {
      int mm = i + ((lane >> 4) << 3);      // D layout: row = i (+8 hi half)
      float sv = sh_s[wave][mm];
      float ir = dr[i] + bir, iz = dz[i] + biz, inn = dn[i] + bin;
      float hr = sv * whr + bhr, hz = sv * whz + bhz, hn = sv * whn + bhn;
      float r = sigm1f(ir + hr);
      float z = sigm1f(iz + hz);
      float nv = tanhf(inn + r * hn);
      float mv = (1.0f - z) * nv + z * sv;
      x1[(size_t)(node0 + mm) * FPDIM + j] = sh_d[wave][mm] * mv;
    }
  }
}

// ------------------------------- layer 1 -----------------------------------
// 2 waves/block (LDS: 2 * 2 * 16*132 floats ~ 33 KB), one tile per wave.
#define LPITCH 132   // padded row pitch (floats) to avoid LDS bank conflicts

__global__ __launch_bounds__(64) void layer1_kernel(
    const float* __restrict__ x1, const int* __restrict__ deg,
    const float* __restrict__ tw, const float* __restrict__ tb,
    const float* __restrict__ wih, const float* __restrict__ whh,
    const float* __restrict__ bih, const float* __restrict__ bhh,
    float* __restrict__ out) {
  __shared__ float shX[2][16 * LPITCH];
  __shared__ float shC[2][16 * LPITCH];
  __shared__ float sh_d[2][16];
  const int lane = threadIdx.x & 31;
  const int wave = threadIdx.x >> 5;
  const int tile = blockIdx.x * 2 + wave;
  const int node0 = tile * 16;
  float* X = &shX[wave][0];
  float* C = &shC[wave][0];

  if (lane < 16) sh_d[wave][lane] = (float)deg[node0 + lane];
  // coalesced tile load: one 128-float row per iteration
#pragma unroll 4
  for (int r = 0; r < 16; ++r) {
    float4 v = *(const float4*)(x1 + (size_t)(node0 + r) * FPDIM + 4 * lane);
    *(float4*)(X + r * LPITCH + 4 * lane) = v;
  }
  __syncthreads();

  const int m  = lane & 15;
  const int kh = (lane >> 4) << 1;

  v2f ax[32];
#pragma unroll
  for (int t = 0; t < 32; ++t)
    ax[t] = *(const v2f*)(X + m * LPITCH + 4 * t + kh);

  // GEMM1: cs = elu(x1 @ tw.T + tb)  -> staged to LDS (D-layout -> A-layout)
#pragma unroll 1
  for (int c = 0; c < 8; ++c) {
    v8f d = {0,0,0,0,0,0,0,0};
#pragma unroll
    for (int t = 0; t < 32; ++t) {
      v2f b = *(const v2f*)(tw + (size_t)(16 * c + m) * FPDIM + 4 * t + kh);
      d = wmma_f32_k4(ax[t], b, d);
    }
    const int j = 16 * c + m;
    const float tbv = tb[j];
#pragma unroll
    for (int i = 0; i < 8; ++i) {
      int mm = i + ((lane >> 4) << 3);
      C[mm * LPITCH + j] = elu1f(d[i] + tbv);
    }
  }
  __syncthreads();

  v2f ac[32];
#pragma unroll
  for (int t = 0; t < 32; ++t)
    ac[t] = *(const v2f*)(C + m * LPITCH + 4 * t + kh);

  // GEMM2: gi = cs @ wih.T (3 gates)  and  gh = x1 @ whh.T (3 gates), fused
#pragma unroll 1
  for (int c = 0; c < 8; ++c) {
    v8f dr = {0,0,0,0,0,0,0,0}, dz = {0,0,0,0,0,0,0,0}, dn = {0,0,0,0,0,0,0,0};
    v8f er = {0,0,0,0,0,0,0,0}, ez = {0,0,0,0,0,0,0,0}, en = {0,0,0,0,0,0,0,0};
#pragma unroll
    for (int t = 0; t < 32; ++t) {
      int k0 = 4 * t + kh;
      v2f b;
      b = *(const v2f*)(wih + (size_t)(16 * c +       m) * FPDIM + k0); dr = wmma_f32_k4(ac[t], b, dr);
      b = *(const v2f*)(wih + (size_t)(16 * c + 128 + m) * FPDIM + k0); dz = wmma_f32_k4(ac[t], b, dz);
      b = *(const v2f*)(wih + (size_t)(16 * c + 256 + m) * FPDIM + k0); dn = wmma_f32_k4(ac[t], b, dn);
      b = *(const v2f*)(whh + (size_t)(16 * c +       m) * FPDIM + k0); er = wmma_f32_k4(ax[t], b, er);
      b = *(const v2f*)(whh + (size_t)(16 * c + 128 + m) * FPDIM + k0); ez = wmma_f32_k4(ax[t], b, ez);
      b = *(const v2f*)(whh + (size_t)(16 * c + 256 + m) * FPDIM + k0); en = wmma_f32_k4(ax[t], b, en);
    }
    const int j = 16 * c + m;
    const float bir = bih[j], biz = bih[j + 128], bin = bih[j + 256];
    const float bhr = bhh[j], bhz = bhh[j + 128], bhn = bhh[j + 256];
#pragma unroll
    for (int i = 0; i < 8; ++i) {
      int mm = i + ((lane >> 4) << 3);
      float ir = dr[i] + bir, iz = dz[i] + biz, inn = dn[i] + bin;
      float hr = er[i] + bhr, hz = ez[i] + bhz, hn = en[i] + bhn;
      float r = sigm1f(ir + hr);
      float z = sigm1f(iz + hz);
      float nv = tanhf(inn + r * hn);
      float h = X[mm * LPITCH + j];
      out[(size_t)(node0 + mm) * FPDIM + j] = sh_d[wave][mm] * ((1.0f - z) * nv + z * h);
    }
  }
}

// ------------------------------- launcher ----------------------------------

extern "C" void kernel_launch(void* const* d_in, const int* in_sizes, int n_in,
                              void* d_out, int out_size, void* d_ws, size_t ws_size,
                              hipStream_t stream) {
  const float* x    = (const float*)d_in[0];
  const int*   edge = (const int*)d_in[1];      // [2, E] flat; row 1 = dst
  // layer 0 params (align_* are mathematically dead: softmax over size-1 axis)
  const float* aw0  = (const float*)d_in[4];    // attend_w0 [128,128]
  const float* ab0  = (const float*)d_in[5];    // attend_b0 [128]
  const float* wih0 = (const float*)d_in[6];
  const float* whh0 = (const float*)d_in[7];
  const float* bih0 = (const float*)d_in[8];
  const float* bhh0 = (const float*)d_in[9];
  // layer 1 params
  const float* aw1  = (const float*)d_in[12];
  const float* ab1  = (const float*)d_in[13];
  const float* wih1 = (const float*)d_in[14];
  const float* whh1 = (const float*)d_in[15];
  const float* bih1 = (const float*)d_in[16];
  const float* bhh1 = (const float*)d_in[17];
  float* out = (float*)d_out;

  // workspace carve-up
  char* ws = (char*)d_ws;
  int*   deg     = (int*)(ws);                         // N ints
  float* sbuf    = (float*)(ws + 819200);              // N floats
  float* twsum0  = (float*)(ws + 819200 + 819200);     // 128 floats
  float* whhsum0 = (float*)(ws + 819200 + 819200 + 4096); // 384 floats
  float* x1      = (float*)(ws + 819200 + 819200 + 8192); // N*128 floats

  zero_i32_kernel<<<(N_NODES + 255) / 256, 256, 0, stream>>>(deg, N_NODES);
  deg_hist_kernel<<<(N_EDGES + 255) / 256, 256, 0, stream>>>(edge + N_EDGES, deg, N_EDGES);
  rowsum_kernel<<<N_NODES / 8, 256, 0, stream>>>(x, sbuf, N_NODES);
  wsum_kernel<<<64, 256, 0, stream>>>(aw0, whh0, twsum0, whhsum0);

  layer0_kernel<<<NTILES / 4, 128, 0, stream>>>(sbuf, deg, twsum0, ab0, whhsum0,
                                                wih0, bih0, bhh0, x1);
  layer1_kernel<<<NTILES / 2, 64, 0, stream>>>(x1, deg, aw1, ab1, wih1, whh1,
                                               bih1, bhh1, out);
}